// MultiHeadAttention_29532195127892
// MI455X (gfx1250) — compile-verified
//
#include <hip/hip_runtime.h>
#include <math.h>

// ---------------------------------------------------------------------------
// Multi-head attention forward, fp32, CDNA5 (gfx1250) using
// V_WMMA_F32_16X16X4_F32 for all four matmuls + flash-style online softmax.
// ---------------------------------------------------------------------------

typedef __attribute__((ext_vector_type(2))) float v2f;
typedef __attribute__((ext_vector_type(8))) float v8f;

#define D_MODEL 512
#define S_LEN   2048
#define N_HEADS 8
#define D_K     64
#define BATCH   2
#define N_ROWS  (BATCH * S_LEN)      // 4096 token rows

static __device__ __forceinline__ v8f wmma_f32(v2f a, v2f b, v8f c) {
  // D = A(16x4,f32) * B(4x16,f32) + C(16x16,f32)
  return __builtin_amdgcn_wmma_f32_16x16x4_f32(
      /*neg_a=*/false, a, /*neg_b=*/false, b,
      /*c_mod=*/(short)0, c, /*reuse_a=*/false, /*reuse_b=*/false);
}

static __device__ __forceinline__ v8f v8f_zero() {
  v8f z = {0.f, 0.f, 0.f, 0.f, 0.f, 0.f, 0.f, 0.f};
  return z;
}

// ---------------------------------------------------------------------------
// Kernel 1: fused QKV projection. y[n,e] = dot(x[n,:], W[e,:]) + bias[e]
// (nn.Linear convention: W is [out,in], y = x @ W^T + b  ->  "NT" gemm)
// One wave computes a 16(M) x 64(N) tile. blockDim = 128 (4 waves).
// grid = (N_ROWS/64, D_MODEL/64, 3), blockIdx.z selects Q / K / V.
// Output layout: [B, H, S, D_K] so the attention kernel reads rows of length 64.
// ---------------------------------------------------------------------------
__global__ void __launch_bounds__(128)
qkv_proj_kernel(const float* __restrict__ x,
                const float* __restrict__ wq, const float* __restrict__ wk,
                const float* __restrict__ wv,
                const float* __restrict__ bq, const float* __restrict__ bk,
                const float* __restrict__ bv,
                float* __restrict__ qo, float* __restrict__ ko,
                float* __restrict__ vo) {
  const int wave  = threadIdx.x >> 5;
  const int lane  = threadIdx.x & 31;
  const int half  = lane >> 4;       // which 16-lane group
  const int col16 = lane & 15;

  const float* w;  const float* bias;  float* out;
  if (blockIdx.z == 0)      { w = wq; bias = bq; out = qo; }
  else if (blockIdx.z == 1) { w = wk; bias = bk; out = ko; }
  else                      { w = wv; bias = bv; out = vo; }

  const int n0 = (blockIdx.x * 4 + wave) * 16;   // row tile base
  const int e0 = blockIdx.y * 64;                // col group base

  v8f acc[4];
#pragma unroll
  for (int t = 0; t < 4; ++t) acc[t] = v8f_zero();

  const float* xrow = x + (size_t)(n0 + col16) * D_MODEL;
  const float* wrow0 = w + (size_t)(e0 +  0 + col16) * D_MODEL;
  const float* wrow1 = w + (size_t)(e0 + 16 + col16) * D_MODEL;
  const float* wrow2 = w + (size_t)(e0 + 32 + col16) * D_MODEL;
  const float* wrow3 = w + (size_t)(e0 + 48 + col16) * D_MODEL;

  for (int k0 = 0; k0 < D_MODEL; k0 += 4) {
    const int kk = k0 + 2 * half;
    v2f a = *(const v2f*)(xrow + kk);       // A frag: x[m][kk..kk+1]
    v2f b0 = *(const v2f*)(wrow0 + kk);     // B frag: W[n][kk..kk+1]
    v2f b1 = *(const v2f*)(wrow1 + kk);
    v2f b2 = *(const v2f*)(wrow2 + kk);
    v2f b3 = *(const v2f*)(wrow3 + kk);
    acc[0] = wmma_f32(a, b0, acc[0]);
    acc[1] = wmma_f32(a, b1, acc[1]);
    acc[2] = wmma_f32(a, b2, acc[2]);
    acc[3] = wmma_f32(a, b3, acc[3]);
  }

  // Epilogue: bias add, scatter to [B,H,S,D_K]
#pragma unroll
  for (int t = 0; t < 4; ++t) {
    const int e  = e0 + t * 16 + col16;
    const float be = bias[e];
    const int h  = e >> 6;        // /64
    const int dk = e & 63;
#pragma unroll
    for (int r = 0; r < 8; ++r) {
      const int n  = n0 + r + 8 * half;   // C layout: row = r + 8*half
      const int b_ = n >> 11;             // / S_LEN (2048)
      const int s  = n & (S_LEN - 1);
      out[(((size_t)b_ * N_HEADS + h) * S_LEN + s) * D_K + dk] = acc[t][r] + be;
    }
  }
}

// ---------------------------------------------------------------------------
// Kernel 2: attention with online softmax (no SxS materialization).
// One wave owns a 16-query tile of one (b,h); loops over 128 key tiles.
// Score tile via WMMA (Q row-major x K row-major, NT). Row stats via shfl_xor
// inside each 16-lane half (matches C-layout row placement M = r + 8*half).
// P transposes C-layout -> A-layout through LDS. P@V via WMMA.
// blockDim = 128 (4 waves = 4 query tiles). grid = (S/64, B*H).
// ---------------------------------------------------------------------------
__global__ void __launch_bounds__(128)
attn_kernel(const float* __restrict__ qw, const float* __restrict__ kw,
            const float* __restrict__ vw, float* __restrict__ ctx) {
  // pad row stride to 18 floats: even (keeps float2 8B alignment), not 16
  // (breaks the 16-float bank periodicity)
  __shared__ __align__(16) float p_lds[4][16][18];

  const int wave  = threadIdx.x >> 5;
  const int lane  = threadIdx.x & 31;
  const int half  = lane >> 4;
  const int col16 = lane & 15;

  const int bh = blockIdx.y;                         // b*H + h
  const int q0 = (blockIdx.x * 4 + wave) * 16;       // query tile base
  const size_t base = (size_t)bh * S_LEN * D_K;

  const float* qrow = qw + base + (size_t)(q0 + col16) * D_K;

  v8f acc[4];
#pragma unroll
  for (int t = 0; t < 4; ++t) acc[t] = v8f_zero();

  float m[8], l[8];
#pragma unroll
  for (int r = 0; r < 8; ++r) { m[r] = -__builtin_huge_valf(); l[r] = 0.f; }

  const float scale = 0.125f;   // 1/sqrt(D_K) = 1/8

  for (int kt = 0; kt < S_LEN / 16; ++kt) {
    const int key0 = kt * 16;
    const float* krow = kw + base + (size_t)(key0 + col16) * D_K;

    // ---- S = (Q K^T) tile, 16 WMMAs over d_k = 64 ----
    v8f sc = v8f_zero();
#pragma unroll
    for (int k0 = 0; k0 < D_K; k0 += 4) {
      const int kk = k0 + 2 * half;
      v2f a = *(const v2f*)(qrow + kk);   // Q[m][kk..kk+1]
      v2f b = *(const v2f*)(krow + kk);   // K[n][kk..kk+1] (B = K^T)
      sc = wmma_f32(a, b, sc);
    }

    // ---- online softmax update (per row r+8*half, spread over 16 lanes) ----
    float p[8];
#pragma unroll
    for (int r = 0; r < 8; ++r) {
      float sv = sc[r] * scale;
      float rm = sv;
      rm = fmaxf(rm, __shfl_xor(rm, 1, 32));
      rm = fmaxf(rm, __shfl_xor(rm, 2, 32));
      rm = fmaxf(rm, __shfl_xor(rm, 4, 32));
      rm = fmaxf(rm, __shfl_xor(rm, 8, 32));
      const float mnew = fmaxf(m[r], rm);
      const float pe = __expf(sv - mnew);
      float rs = pe;
      rs += __shfl_xor(rs, 1, 32);
      rs += __shfl_xor(rs, 2, 32);
      rs += __shfl_xor(rs, 4, 32);
      rs += __shfl_xor(rs, 8, 32);
      const float sf = __expf(m[r] - mnew);
      l[r] = l[r] * sf + rs;
      m[r] = mnew;
      p[r] = pe;
      acc[0][r] *= sf;
      acc[1][r] *= sf;
      acc[2][r] *= sf;
      acc[3][r] *= sf;
    }

    // ---- transpose P (C layout) -> A layout through LDS ----
#pragma unroll
    for (int r = 0; r < 8; ++r)
      p_lds[wave][r + 8 * half][col16] = p[r];
    __syncthreads();

    // ---- O += P @ V (K = 16 keys -> 4 k-steps, 4 d-tiles) ----
#pragma unroll
    for (int k0 = 0; k0 < 16; k0 += 4) {
      const int kk = k0 + 2 * half;
      v2f a = *(const v2f*)&p_lds[wave][col16][kk];   // P[q][kk..kk+1]
      const float* v0 = vw + base + (size_t)(key0 + kk) * D_K + col16;
#pragma unroll
      for (int t = 0; t < 4; ++t) {
        v2f b;
        b.x = v0[t * 16];          // V[key0+kk  ][d]
        b.y = v0[t * 16 + D_K];    // V[key0+kk+1][d]
        acc[t] = wmma_f32(a, b, acc[t]);
      }
    }
    __syncthreads();   // protect p_lds before next tile overwrites it
  }

  // ---- normalize by l and write context [B, S, D_MODEL] ----
  const int b_ = bh >> 3;   // / N_HEADS
  const int h  = bh & 7;
#pragma unroll
  for (int r = 0; r < 8; ++r) {
    const float inv = 1.0f / l[r];
    const int sidx = q0 + r + 8 * half;
#pragma unroll
    for (int t = 0; t < 4; ++t) {
      const int e = h * D_K + t * 16 + col16;
      ctx[((size_t)b_ * S_LEN + sidx) * D_MODEL + e] = acc[t][r] * inv;
    }
  }
}

// ---------------------------------------------------------------------------
// Kernel 3: output projection. out[n,e] = dot(ctx[n,:], Wo[e,:]) + bo[e]
// Same tiling as kernel 1, dense [B,S,D_MODEL] output.
// ---------------------------------------------------------------------------
__global__ void __launch_bounds__(128)
out_proj_kernel(const float* __restrict__ ctx, const float* __restrict__ wo,
                const float* __restrict__ bo, float* __restrict__ out) {
  const int wave  = threadIdx.x >> 5;
  const int lane  = threadIdx.x & 31;
  const int half  = lane >> 4;
  const int col16 = lane & 15;

  const int n0 = (blockIdx.x * 4 + wave) * 16;
  const int e0 = blockIdx.y * 64;

  v8f acc[4];
#pragma unroll
  for (int t = 0; t < 4; ++t) acc[t] = v8f_zero();

  const float* xrow = ctx + (size_t)(n0 + col16) * D_MODEL;
  const float* wrow0 = wo + (size_t)(e0 +  0 + col16) * D_MODEL;
  const float* wrow1 = wo + (size_t)(e0 + 16 + col16) * D_MODEL;
  const float* wrow2 = wo + (size_t)(e0 + 32 + col16) * D_MODEL;
  const float* wrow3 = wo + (size_t)(e0 + 48 + col16) * D_MODEL;

  for (int k0 = 0; k0 < D_MODEL; k0 += 4) {
    const int kk = k0 + 2 * half;
    v2f a  = *(const v2f*)(xrow + kk);
    v2f b0 = *(const v2f*)(wrow0 + kk);
    v2f b1 = *(const v2f*)(wrow1 + kk);
    v2f b2 = *(const v2f*)(wrow2 + kk);
    v2f b3 = *(const v2f*)(wrow3 + kk);
    acc[0] = wmma_f32(a, b0, acc[0]);
    acc[1] = wmma_f32(a, b1, acc[1]);
    acc[2] = wmma_f32(a, b2, acc[2]);
    acc[3] = wmma_f32(a, b3, acc[3]);
  }

#pragma unroll
  for (int t = 0; t < 4; ++t) {
    const int e = e0 + t * 16 + col16;
    const float be = bo[e];
#pragma unroll
    for (int r = 0; r < 8; ++r) {
      const int n = n0 + r + 8 * half;
      out[(size_t)n * D_MODEL + e] = acc[t][r] + be;
    }
  }
}

// ---------------------------------------------------------------------------
extern "C" void kernel_launch(void* const* d_in, const int* in_sizes, int n_in,
                              void* d_out, int out_size, void* d_ws,
                              size_t ws_size, hipStream_t stream) {
  const float* x  = (const float*)d_in[0];
  const float* wq = (const float*)d_in[1];
  const float* bq = (const float*)d_in[2];
  const float* wk = (const float*)d_in[3];
  const float* bk = (const float*)d_in[4];
  const float* wv = (const float*)d_in[5];
  const float* bv = (const float*)d_in[6];
  const float* wo = (const float*)d_in[7];
  const float* bo = (const float*)d_in[8];
  float* out = (float*)d_out;

  // workspace: Q, K, V in [B,H,S,D_K] + context [B,S,D_MODEL]  (32 MB fp32)
  float* ws = (float*)d_ws;
  const size_t qkv_elems = (size_t)BATCH * N_HEADS * S_LEN * D_K;  // 2M floats
  float* qws = ws;
  float* kws = qws + qkv_elems;
  float* vws = kws + qkv_elems;
  float* ctx = vws + qkv_elems;

  qkv_proj_kernel<<<dim3(N_ROWS / 64, D_MODEL / 64, 3), 128, 0, stream>>>(
      x, wq, wk, wv, bq, bk, bv, qws, kws, vws);

  attn_kernel<<<dim3(S_LEN / 64, BATCH * N_HEADS), 128, 0, stream>>>(
      qws, kws, vws, ctx);

  out_proj_kernel<<<dim3(N_ROWS / 64, D_MODEL / 64), 128, 0, stream>>>(
      ctx, wo, bo, out);
}